// GAT_10033043604233
// MI455X (gfx1250) — compile-verified
//
#include <hip/hip_runtime.h>
#include <math.h>

typedef float v2f __attribute__((ext_vector_type(2)));
typedef float v8f __attribute__((ext_vector_type(8)));

#define IN_FT   128
#define OUT_FT  64
#define HEADS   2
#define FT      (HEADS * OUT_FT)   /* 128 */
#define NEG_SLOPE 0.2f

__device__ __forceinline__ float leaky(float s) {
    return s > 0.f ? s : NEG_SLOPE * s;
}

// ---------------------------------------------------------------------------
// K0: zero output, init per-(node,head) running max to -inf and denom to 0
// ---------------------------------------------------------------------------
__global__ void gat_init_kernel(float* __restrict__ out,
                                float* __restrict__ mmax,
                                float* __restrict__ den,
                                int n_out, int n_nh) {
    int i = blockIdx.x * blockDim.x + threadIdx.x;
    if (i < n_out) out[i] = 0.f;
    if (i < n_nh) { mmax[i] = -INFINITY; den[i] = 0.f; }
}

// ---------------------------------------------------------------------------
// K1: xt = x @ W via V_WMMA_F32_16X16X4_F32.
// Block = 128 threads = 4 waves. All 4 waves share one 16-column slice of W
// (staged into LDS, 8 KB); each wave computes one 16x16 output tile.
//   A 16x4 f32 frag: lanes 0-15 hold M=lane {K=k0,k0+1}; lanes 16-31 {K=k0+2,k0+3}
//   B 4x16 f32 frag: lanes 0-15 hold N=lane rows K=k0,k0+1; lanes 16-31 K=k0+2,k0+3
//   C/D 16x16 f32:   VGPR r -> M=r (lanes 0-15, N=lane) / M=r+8 (lanes 16-31)
// ---------------------------------------------------------------------------
__global__ __launch_bounds__(128)
void gat_gemm_wmma_kernel(const float* __restrict__ x,
                          const float* __restrict__ W,
                          float* __restrict__ xt, int N) {
    __shared__ float Wl[IN_FT * 16];

    const int tile_n = blockIdx.x & 7;         // 8 column tiles of 16 (FT=128)
    const int group  = blockIdx.x >> 3;        // group of 4 row tiles
    const int col0   = tile_n * 16;

    // stage W[:, col0:col0+16] into LDS
    for (int i = threadIdx.x; i < IN_FT * 16; i += 128) {
        int k = i >> 4, c = i & 15;
        Wl[i] = W[k * FT + col0 + c];
    }
    __syncthreads();

    const int wave = threadIdx.x >> 5;
    const int lane = threadIdx.x & 31;
    const int half = lane >> 4;                // 0: lanes 0-15, 1: lanes 16-31
    const int l16  = lane & 15;

    const int row0 = (group * 4 + wave) * 16;
    int arow = row0 + l16;
    if (arow >= N) arow = N - 1;               // clamp; stores are guarded

    v8f c = {};
    for (int k0 = 0; k0 < IN_FT; k0 += 4) {
        v2f a, b;
        const float* ap = x + (size_t)arow * IN_FT + (k0 + 2 * half);
        a.x = ap[0];                           // K = k0 + 2*half
        a.y = ap[1];                           // K = k0 + 2*half + 1
        b.x = Wl[(k0 + 2 * half)     * 16 + l16];
        b.y = Wl[(k0 + 2 * half + 1) * 16 + l16];
        // (neg_a, A, neg_b, B, c_mod, C, reuse_a, reuse_b)
        c = __builtin_amdgcn_wmma_f32_16x16x4_f32(false, a, false, b,
                                                  (short)0, c, false, false);
    }

    #pragma unroll
    for (int r = 0; r < 8; ++r) {
        int rr = row0 + r + 8 * half;
        if (rr < N) xt[(size_t)rr * FT + col0 + l16] = c[r];
    }
}

// ---------------------------------------------------------------------------
// K2: per-(node,head) attention logits: a_src/a_dst = <xt[n,h,:], att_*[h,:]>
// ---------------------------------------------------------------------------
__global__ void gat_logits_kernel(const float* __restrict__ xt,
                                  const float* __restrict__ att_src,
                                  const float* __restrict__ att_dst,
                                  float* __restrict__ a_src,
                                  float* __restrict__ a_dst, int NH) {
    int i = blockIdx.x * blockDim.x + threadIdx.x;
    if (i >= NH) return;
    int n = i >> 1, h = i & 1;
    const float4* xp = (const float4*)(xt + (size_t)n * FT + h * OUT_FT);
    const float4* as = (const float4*)(att_src + h * OUT_FT);
    const float4* ad = (const float4*)(att_dst + h * OUT_FT);
    float s0 = 0.f, s1 = 0.f;
    #pragma unroll
    for (int j = 0; j < OUT_FT / 4; ++j) {
        float4 v = xp[j], a = as[j], d = ad[j];
        s0 += v.x * a.x + v.y * a.y + v.z * a.z + v.w * a.w;
        s1 += v.x * d.x + v.y * d.y + v.z * d.z + v.w * d.w;
    }
    a_src[i] = s0;
    a_dst[i] = s1;
}

// ---------------------------------------------------------------------------
// K3: segment max over incoming edges (float atomicMax)
// ---------------------------------------------------------------------------
__global__ void gat_edge_max_kernel(const int* __restrict__ ei,
                                    const float* __restrict__ a_src,
                                    const float* __restrict__ a_dst,
                                    float* __restrict__ mmax, int E) {
    int e = blockIdx.x * blockDim.x + threadIdx.x;
    if (e >= E) return;
    int s = ei[e], d = ei[E + e];
    #pragma unroll
    for (int h = 0; h < HEADS; ++h) {
        float v = leaky(a_src[s * HEADS + h] + a_dst[d * HEADS + h]);
        atomicMax(&mmax[d * HEADS + h], v);
    }
}

// ---------------------------------------------------------------------------
// K4: segment exp-sum (recompute score; float atomicAdd)
// ---------------------------------------------------------------------------
__global__ void gat_edge_expsum_kernel(const int* __restrict__ ei,
                                       const float* __restrict__ a_src,
                                       const float* __restrict__ a_dst,
                                       const float* __restrict__ mmax,
                                       float* __restrict__ den, int E) {
    int e = blockIdx.x * blockDim.x + threadIdx.x;
    if (e >= E) return;
    int s = ei[e], d = ei[E + e];
    #pragma unroll
    for (int h = 0; h < HEADS; ++h) {
        float v = leaky(a_src[s * HEADS + h] + a_dst[d * HEADS + h]);
        atomicAdd(&den[d * HEADS + h], __expf(v - mmax[d * HEADS + h]));
    }
}

// ---------------------------------------------------------------------------
// K5: one wave per edge; lane l handles features [4l, 4l+4).
//     lanes 0-15 -> head 0, lanes 16-31 -> head 1.
//     global_load_b128 gather of xt[src], 4x atomic f32 add into out[dst].
// ---------------------------------------------------------------------------
__global__ __launch_bounds__(256)
void gat_edge_scatter_kernel(const int* __restrict__ ei,
                             const float* __restrict__ a_src,
                             const float* __restrict__ a_dst,
                             const float* __restrict__ mmax,
                             const float* __restrict__ den,
                             const float* __restrict__ xt,
                             float* __restrict__ out, int E) {
    int gid  = blockIdx.x * blockDim.x + threadIdx.x;
    int e    = gid >> 5;
    int lane = gid & 31;
    if (e >= E) return;
    int s = ei[e], d = ei[E + e];
    int h = lane >> 4;
    float v    = leaky(a_src[s * HEADS + h] + a_dst[d * HEADS + h]);
    float coef = __expf(v - mmax[d * HEADS + h]) / den[d * HEADS + h];
    float4 xv = ((const float4*)(xt + (size_t)s * FT))[lane];
    float* op = out + (size_t)d * FT + lane * 4;
    atomicAdd(op + 0, coef * xv.x);
    atomicAdd(op + 1, coef * xv.y);
    atomicAdd(op + 2, coef * xv.z);
    atomicAdd(op + 3, coef * xv.w);
}

// ---------------------------------------------------------------------------
// K6: bias + PReLU epilogue
// ---------------------------------------------------------------------------
__global__ void gat_epilogue_kernel(float* __restrict__ out,
                                    const float* __restrict__ bias,
                                    const float* __restrict__ prelu_w, int total) {
    int i = blockIdx.x * blockDim.x + threadIdx.x;
    if (i >= total) return;
    float v = out[i] + bias[i & (FT - 1)];
    out[i] = v >= 0.f ? v : prelu_w[0] * v;
}

// ---------------------------------------------------------------------------
extern "C" void kernel_launch(void* const* d_in, const int* in_sizes, int n_in,
                              void* d_out, int out_size, void* d_ws, size_t ws_size,
                              hipStream_t stream) {
    const float* x       = (const float*)d_in[0];
    const int*   ei      = (const int*)  d_in[1];
    const float* W       = (const float*)d_in[2];
    const float* att_src = (const float*)d_in[3];
    const float* att_dst = (const float*)d_in[4];
    const float* bias    = (const float*)d_in[5];
    const float* prelu_w = (const float*)d_in[6];

    const int N = in_sizes[0] / IN_FT;   // 50000
    const int E = in_sizes[1] / 2;       // 800000

    float* out = (float*)d_out;

    // workspace layout (floats)
    float* ws    = (float*)d_ws;
    float* xt    = ws;                             // N*FT
    float* a_src = xt    + (size_t)N * FT;         // N*HEADS
    float* a_dst = a_src + (size_t)N * HEADS;      // N*HEADS
    float* mmax  = a_dst + (size_t)N * HEADS;      // N*HEADS
    float* den   = mmax  + (size_t)N * HEADS;      // N*HEADS

    const int NH    = N * HEADS;
    const int total = N * FT;

    gat_init_kernel<<<(total + 255) / 256, 256, 0, stream>>>(out, mmax, den, total, NH);

    const int tiles_m = (N + 15) / 16;
    const int groups  = (tiles_m + 3) / 4;
    gat_gemm_wmma_kernel<<<groups * 8, 128, 0, stream>>>(x, W, xt, N);

    gat_logits_kernel<<<(NH + 255) / 256, 256, 0, stream>>>(xt, att_src, att_dst,
                                                            a_src, a_dst, NH);

    gat_edge_max_kernel<<<(E + 255) / 256, 256, 0, stream>>>(ei, a_src, a_dst, mmax, E);
    gat_edge_expsum_kernel<<<(E + 255) / 256, 256, 0, stream>>>(ei, a_src, a_dst,
                                                                mmax, den, E);

    long long sc_threads = (long long)E * 32;
    gat_edge_scatter_kernel<<<(unsigned)((sc_threads + 255) / 256), 256, 0, stream>>>(
        ei, a_src, a_dst, mmax, den, xt, out, E);

    gat_epilogue_kernel<<<(total + 255) / 256, 256, 0, stream>>>(out, bias, prelu_w, total);
}